// tsail_sur_14250701488883
// MI455X (gfx1250) — compile-verified
//
#include <hip/hip_runtime.h>

typedef __attribute__((ext_vector_type(2))) float v2f;
typedef __attribute__((ext_vector_type(8))) float v8f;

// ---------------------------------------------------------------------------
// Tall-skinny fp32 GEMM:  C[M,F] = A[M,K] @ W[K,F] + bias, optional ReLU.
// Requires M % 16 == 0 and K % 4 == 0 (true for all layers here).
// One wave computes a 16x32 tile (two 16x16 WMMA tiles sharing the A
// fragment) via V_WMMA_F32_16X16X4_F32.
//
// Fragment layouts per CDNA5 ISA 7.12.2 (32-bit A 16x4 / B 4x16 / C 16x16):
//   A: lane<16 -> M=lane, holds K = {0,1}; lane>=16 -> M=lane-16, K = {2,3}
//   B: lane<16 -> N=lane, holds K = {0,1}; lane>=16 -> N=lane-16, K = {2,3}
//   C/D: VGPR r -> M = r + 8*(lane>=16), N = lane&15
//
// Out-of-range columns are handled by clamping the load address (always
// in-bounds -> unconditional global_load_b32) and zeroing via v_cndmask,
// so the k-loop is branch-free and EXEC stays all-1s for WMMA.
// ---------------------------------------------------------------------------
__global__ __launch_bounds__(128) void gemm_wmma_f32(
    const float* __restrict__ A, const float* __restrict__ W,
    const float* __restrict__ bias, float* __restrict__ C,
    int M, int K, int F, int relu)
{
  const int lane  = threadIdx.x & 31;
  const int wave  = threadIdx.x >> 5;
  const int mtile = blockIdx.x * 4 + wave;
  if (mtile * 16 >= M) return;                 // wave-uniform exit
  const int kh   = lane >> 4;                  // 0: lanes 0-15, 1: lanes 16-31
  const int row  = mtile * 16 + (lane & 15);
  const int col0 = blockIdx.y * 32 + (lane & 15);
  const int col1 = col0 + 16;
  const bool cok0 = (col0 < F);
  const bool cok1 = (col1 < F);
  const int  cc0  = cok0 ? col0 : (F - 1);     // clamped -> loads never fault
  const int  cc1  = cok1 ? col1 : (F - 1);

  v8f acc0 = {0.f, 0.f, 0.f, 0.f, 0.f, 0.f, 0.f, 0.f};
  v8f acc1 = {0.f, 0.f, 0.f, 0.f, 0.f, 0.f, 0.f, 0.f};
  const float* arow = A + (size_t)row * K;
  const float* w0   = W + cc0;
  const float* w1   = W + cc1;

  for (int k = 0; k < K; k += 4) {
    const int ka = k + 2 * kh;
    v2f a = *(const v2f*)(arow + ka);          // K even, ka even -> 8B aligned
    const float bx0 = w0[(size_t)ka * F];
    const float by0 = w0[(size_t)(ka + 1) * F];
    const float bx1 = w1[(size_t)ka * F];
    const float by1 = w1[(size_t)(ka + 1) * F];
    v2f b0, b1;
    b0.x = cok0 ? bx0 : 0.f;  b0.y = cok0 ? by0 : 0.f;
    b1.x = cok1 ? bx1 : 0.f;  b1.y = cok1 ? by1 : 0.f;
    acc0 = __builtin_amdgcn_wmma_f32_16x16x4_f32(
        false, a, false, b0, (short)0, acc0, false, false);
    acc1 = __builtin_amdgcn_wmma_f32_16x16x4_f32(
        false, a, false, b1, (short)0, acc1, false, false);
  }

  const float bb0 = (cok0 && bias != nullptr) ? bias[col0] : 0.f;
  const float bb1 = (cok1 && bias != nullptr) ? bias[col1] : 0.f;
#pragma unroll
  for (int r = 0; r < 8; ++r) {
    const int orow = mtile * 16 + r + 8 * kh;
    float v0 = acc0[r] + bb0;
    float v1 = acc1[r] + bb1;
    if (relu) { v0 = fmaxf(v0, 0.f); v1 = fmaxf(v1, 0.f); }
    float* crow = C + (size_t)orow * F;
    if (cok0) crow[col0] = v0;
    if (cok1) crow[col1] = v1;
  }
}

// ---------------------------------------------------------------------------
// SpMM scatter: out[row[e], :] += val[e] * h[col[e], :]   (out pre-zeroed)
// One wave per edge; lanes stride the feature dim; native fp32 atomics.
// Gathers/scatters stay resident in the 192MB L2 (h is <= 80MB).
// ---------------------------------------------------------------------------
__global__ __launch_bounds__(256) void spmm_scatter(
    const int* __restrict__ er, const int* __restrict__ ec,
    const float* __restrict__ ev, const float* __restrict__ h,
    float* __restrict__ out, int E, int F)
{
  const int gw   = (int)((blockIdx.x * 256u + threadIdx.x) >> 5);
  const int lane = threadIdx.x & 31;
  if (gw >= E) return;
  const int   r = er[gw];
  const int   c = ec[gw];
  const float v = ev[gw];
  const float* hrow = h + (size_t)c * F;
  float*       orow = out + (size_t)r * F;
  for (int f = lane; f < F; f += 32)
    unsafeAtomicAdd(orow + f, v * hrow[f]);
}

// ---------------------------------------------------------------------------
// Elementwise helpers
// ---------------------------------------------------------------------------
__global__ void fill_zero(float* __restrict__ p, size_t n)
{
  size_t i = (size_t)blockIdx.x * blockDim.x + threadIdx.x;
  const size_t stride = (size_t)gridDim.x * blockDim.x;
  for (; i < n; i += stride) p[i] = 0.f;
}

__global__ void add_relu(const float* __restrict__ a, const float* __restrict__ b,
                         float* __restrict__ o, size_t n)
{
  size_t i = (size_t)blockIdx.x * blockDim.x + threadIdx.x;
  const size_t stride = (size_t)gridDim.x * blockDim.x;
  for (; i < n; i += stride) o[i] = fmaxf(a[i] + b[i], 0.f);
}

// ---------------------------------------------------------------------------
// BatchNorm (training mode, biased variance) over N rows x 256 cols.
// Pass 1: column-per-thread register accumulation, atomic combine.
// ---------------------------------------------------------------------------
__global__ __launch_bounds__(256) void bn_stats(
    const float* __restrict__ x, float* __restrict__ sums,
    float* __restrict__ sqs, int Nrows)
{
  const int c = threadIdx.x;          // 256 threads == 256 columns
  float s = 0.f, q = 0.f;
  for (int r = blockIdx.x; r < Nrows; r += gridDim.x) {
    const float v = x[(size_t)r * 256 + c];
    s += v;
    q += v * v;
  }
  unsafeAtomicAdd(&sums[c], s);
  unsafeAtomicAdd(&sqs[c], q);
}

__global__ void bn_finalize(const float* __restrict__ sums, const float* __restrict__ sqs,
                            const float* __restrict__ gamma, const float* __restrict__ beta,
                            float* __restrict__ scale, float* __restrict__ shift,
                            float invN)
{
  const int c = threadIdx.x;
  const float mu  = sums[c] * invN;
  const float var = sqs[c] * invN - mu * mu;
  const float a   = gamma[c] * rsqrtf(var + 1e-5f);
  scale[c] = a;
  shift[c] = beta[c] - mu * a;
}

__global__ void bn_apply_relu(float* __restrict__ x, const float* __restrict__ scale,
                              const float* __restrict__ shift, size_t n)
{
  size_t i = (size_t)blockIdx.x * blockDim.x + threadIdx.x;
  const size_t stride = (size_t)gridDim.x * blockDim.x;
  for (; i < n; i += stride) {
    const int c = (int)(i & 255u);
    x[i] = fmaxf(x[i] * scale[c] + shift[c], 0.f);
  }
}

// ---------------------------------------------------------------------------
// Host orchestration
// ---------------------------------------------------------------------------
extern "C" void kernel_launch(void* const* d_in, const int* in_sizes, int n_in,
                              void* d_out, int out_size, void* d_ws, size_t ws_size,
                              hipStream_t stream)
{
  const float* x     = (const float*)d_in[0];
  const int*   er    = (const int*)d_in[1];
  const int*   ec    = (const int*)d_in[2];
  const float* ev    = (const float*)d_in[3];
  const float* W0    = (const float*)d_in[4];
  const float* b0    = (const float*)d_in[5];
  const float* Ws0   = (const float*)d_in[6];
  const float* bs0   = (const float*)d_in[7];
  const float* W1    = (const float*)d_in[8];
  const float* b1    = (const float*)d_in[9];
  const float* Ws1   = (const float*)d_in[10];
  const float* bs1   = (const float*)d_in[11];
  const float* W2    = (const float*)d_in[12];
  const float* b2    = (const float*)d_in[13];
  const float* Ws2   = (const float*)d_in[14];
  const float* bs2   = (const float*)d_in[15];
  const float* Wfc1  = (const float*)d_in[16];
  const float* bfc1  = (const float*)d_in[17];
  const float* Wfc2a = (const float*)d_in[18];
  const float* bfc2a = (const float*)d_in[19];
  const float* gam   = (const float*)d_in[20];
  const float* bet   = (const float*)d_in[21];
  const float* Wfc2b = (const float*)d_in[22];
  const float* bfc2b = (const float*)d_in[23];

  const int N = in_sizes[0] / 100;   // 100000 (multiple of 16)
  const int E = in_sizes[1];         // 1600000

  // Workspace layout (floats): B0[N*200] B1[N*200] B2[N*256] B3[N*200] stats[1024]
  float* B0    = (float*)d_ws;
  float* B1    = B0 + (size_t)N * 200;
  float* B2    = B1 + (size_t)N * 200;
  float* B3    = B2 + (size_t)N * 256;
  float* sums  = B3 + (size_t)N * 200;
  float* sqs   = sums + 256;
  float* scale = sqs + 256;
  float* shift = scale + 256;

  auto gemm = [&](const float* Ain, const float* Wm, const float* bv,
                  float* Cout, int K, int F, int relu) {
    dim3 grid((unsigned)((N / 16 + 3) / 4), (unsigned)((F + 31) / 32));
    gemm_wmma_f32<<<grid, 128, 0, stream>>>(Ain, Wm, bv, Cout, N, K, F, relu);
  };
  auto zero = [&](float* p, size_t n) {
    fill_zero<<<2048, 256, 0, stream>>>(p, n);
  };
  auto spmm = [&](const float* h, float* s, int F) {
    spmm_scatter<<<(unsigned)((E + 7) / 8), 256, 0, stream>>>(er, ec, ev, h, s, E, F);
  };
  auto arelu = [&](const float* a, const float* b, float* o, size_t n) {
    add_relu<<<2048, 256, 0, stream>>>(a, b, o, n);
  };

  // ---- Layer 0: 100 -> 200 ----
  gemm(x, W0,  b0,  B0, 100, 200, 0);            // h  = x@W0 + b0
  gemm(x, Ws0, bs0, B1, 100, 200, 0);            // x2 = x@Ws0 + bs0
  zero(B2, (size_t)N * 200);
  spmm(B0, B2, 200);                             // s  = A @ h
  arelu(B2, B1, B3, (size_t)N * 200);            // x  = relu(s + x2) -> B3

  // ---- Layer 1: 200 -> 128 ----
  gemm(B3, W1,  b1,  B0, 200, 128, 0);           // h  -> B0
  gemm(B3, Ws1, bs1, B1, 200, 128, 0);           // x2 -> B1
  zero(B2, (size_t)N * 128);
  spmm(B0, B2, 128);                             // s  -> B2
  arelu(B2, B1, B0, (size_t)N * 128);            // x  -> B0

  // ---- Layer 2: 128 -> 128 ----
  gemm(B0, W2,  b2,  B1, 128, 128, 0);           // h  -> B1
  gemm(B0, Ws2, bs2, B2, 128, 128, 0);           // x2 -> B2
  zero(B3, (size_t)N * 128);
  spmm(B1, B3, 128);                             // s  -> B3
  arelu(B3, B2, B1, (size_t)N * 128);            // x  -> B1

  // ---- MLP head ----
  gemm(B1, Wfc1,  bfc1,  B0, 128, 128, 1);       // y1 = relu(x@Wfc1+bfc1) -> B0
  gemm(B0, Wfc2a, bfc2a, B2, 128, 256, 0);       // y2 = y1@Wfc2a+bfc2a    -> B2

  // BatchNorm over batch (biased var) + ReLU, in-place on B2
  zero(sums, 512);                               // sums + sqs contiguous
  bn_stats<<<1024, 256, 0, stream>>>(B2, sums, sqs, N);
  bn_finalize<<<1, 256, 0, stream>>>(sums, sqs, gam, bet, scale, shift, 1.0f / (float)N);
  bn_apply_relu<<<2048, 256, 0, stream>>>(B2, scale, shift, (size_t)N * 256);

  // Output: 256 -> 18
  gemm(B2, Wfc2b, bfc2b, (float*)d_out, 256, 18, 0);
}